// BallQueryLayer_66846870995165
// MI455X (gfx1250) — compile-verified
//
#include <hip/hip_runtime.h>

typedef __attribute__((ext_vector_type(2))) float v2f;
typedef __attribute__((ext_vector_type(8))) float v8f;

#define NPTS 65536
#define KNN 32
#define RADIUS2 (0.05f * 0.05f)
#define NT (NPTS / 16)

// ---------------------------------------------------------------------------
// Zero-init: mapping (int32, default 0), num_neighbors, outputs (0.0f) all
// share bit pattern 0x00000000, so one vectorized memset covers d_out.
// ---------------------------------------------------------------------------
__global__ __launch_bounds__(256) void bq_zero_kernel(float* __restrict__ out, int n4) {
    float4 z = make_float4(0.f, 0.f, 0.f, 0.f);
    for (int i = blockIdx.x * blockDim.x + threadIdx.x; i < n4;
         i += gridDim.x * blockDim.x)
        ((float4*)out)[i] = z;
}

// ---------------------------------------------------------------------------
// One wave32 owns TWO tiles of 16 queries (32 queries); streams points2 in
// 16-point tiles with a 2-deep ping-pong pipeline (no register rotation).
// Per B tile, TWO v_wmma_f32_16x16x4_f32 produce both 16x16 d^2 tiles:
//   A row = (-2qx,-2qy,-2qz,1), B col = (px,py,pz,|p|^2), C = |q|^2 (invariant)
//   D = |q|^2 + |p|^2 - 2 q.p = d^2
// Fast path: 1 b96 load, 3 FMA, 2 cndmask, 2 WMMA, v_min3 tree + 1 ballot.
// Hit path (~23% of tiles): per-row ballots, mbcnt-prefix deterministic
// ascending slots, scalar counters, direct scatter of mapping/outputs.
// ---------------------------------------------------------------------------
__global__ __launch_bounds__(256, 1) void bq_main_kernel(
    const float* __restrict__ q, const float* __restrict__ p,
    int* __restrict__ mapOut, int* __restrict__ nnOut,
    float* __restrict__ outOut) {

    const int lane = threadIdx.x & 31;
    const int w    = blockIdx.x * 8 + (threadIdx.x >> 5);  // wave id 0..2047
    const int qt0  = w * 2;                                // query tiles qt0, qt0+1
    const int m    = lane & 15;
    const int hi   = lane >> 4;

    // ---- query setup (once): A (scaled by -2, K3=1) and C = |q|^2 ---------
    v2f A0, A1;
    v8f C0, C1;
    {
        int qidx = qt0 * 16 + m;
        float qx = q[qidx * 3 + 0], qy = q[qidx * 3 + 1], qz = q[qidx * 3 + 2];
        float q2 = qx * qx + qy * qy + qz * qz;
        A0.x = hi ? (-2.0f * qz) : (-2.0f * qx);
        A0.y = hi ? 1.0f         : (-2.0f * qy);
#pragma unroll
        for (int v = 0; v < 8; ++v) C0[v] = __shfl(q2, v + 8 * hi, 32);
    }
    {
        int qidx = (qt0 + 1) * 16 + m;
        float qx = q[qidx * 3 + 0], qy = q[qidx * 3 + 1], qz = q[qidx * 3 + 2];
        float q2 = qx * qx + qy * qy + qz * qz;
        A1.x = hi ? (-2.0f * qz) : (-2.0f * qx);
        A1.y = hi ? 1.0f         : (-2.0f * qy);
#pragma unroll
        for (int v = 0; v < 8; ++v) C1[v] = __shfl(q2, v + 8 * hi, 32);
    }

    unsigned cnt0[16], cnt1[16];  // uniform (SGPR) per-query counters
#pragma unroll
    for (int i = 0; i < 16; ++i) { cnt0[i] = 0u; cnt1[i] = 0u; }

    const unsigned halfsel = hi ? 0xFFFF0000u : 0x0000FFFFu;

    // append hits of one d^2 tile (deterministic ascending-index slots)
    auto append = [&](const v8f& D, unsigned (&cnt)[16], int qtb, int pidx,
                      float f0, float f1, float f2) {
#pragma unroll
        for (int v = 0; v < 8; ++v) {
            unsigned msk = (unsigned)__builtin_amdgcn_ballot_w32(D[v] <= RADIUS2);
            if (msk == 0u) continue;  // uniform skip
            bool active  = (msk >> lane) & 1u;
            int prefix   = (int)__builtin_amdgcn_mbcnt_lo(msk & halfsel, 0u);
            unsigned bc  = hi ? cnt[v + 8] : cnt[v];
            int slot     = (int)bc + prefix;
            if (active && slot < KNN) {
                int qg = qtb * 16 + v + 8 * hi;
                mapOut[qg * KNN + slot] = pidx;
                float* o = outOut + (qg * KNN + slot) * 3;
                o[0] = f0; o[1] = f1; o[2] = f2;  // lane's own point coords
            }
            cnt[v]     += (unsigned)__popc(msk & 0xFFFFu);
            cnt[v + 8] += (unsigned)__popc(msk >> 16);
        }
    };

    // process one 16-point tile against both query tiles
    auto process = [&](int pt, float f0, float f1, float f2) {
        float sq = f0 * f0 + f1 * f1 + f2 * f2;     // |p|^2 (hi lanes)
        v2f B;
        B.x = hi ? f2 : f0;
        B.y = hi ? sq : f1;
        v8f D0 = __builtin_amdgcn_wmma_f32_16x16x4_f32(
            false, A0, false, B, (short)0, C0, false, false);
        v8f D1 = __builtin_amdgcn_wmma_f32_16x16x4_f32(
            false, A1, false, B, (short)0, C1, false, false);
        // v_min3 tree over the 16 d^2 registers -> single any-hit ballot
        float t0 = fminf(fminf(D0[0], D0[1]), D0[2]);
        float t1 = fminf(fminf(D0[3], D0[4]), D0[5]);
        float t2 = fminf(fminf(D0[6], D0[7]), D1[0]);
        float t3 = fminf(fminf(D1[1], D1[2]), D1[3]);
        float t4 = fminf(fminf(D1[4], D1[5]), D1[6]);
        float u0 = fminf(fminf(t0, t1), D1[7]);
        float u1 = fminf(fminf(t2, t3), t4);
        float mn = fminf(u0, u1);
        if (__builtin_amdgcn_ballot_w32(mn <= RADIUS2)) {  // uniform branch
            int pidx = pt * 16 + m;
            append(D0, cnt0, qt0,     pidx, f0, f1, f2);
            append(D1, cnt1, qt0 + 1, pidx, f0, f1, f2);
        }
    };

    // ---- 2-stage ping-pong pipeline over all 4096 point tiles -------------
    {
        const int moff = m * 3;
        int tbase = 0;  // uniform float index of current tile pt
        float a0 = p[moff + 0], a1 = p[moff + 1], a2 = p[moff + 2];
        float b0, b1, b2;
        int pt = 0;
        for (; pt < NT - 2; pt += 2) {
            // WGP-scope prefetch (default scope 0 -> fills WGP$/L0), 8 tiles
            // ahead; lane*12 spread covers the full 384B window; clamp keeps
            // the non-speculative prefetch in-bounds.
            {
                int pfo = tbase + 8 * 48 + lane * 3;
                pfo = pfo < (3 * NPTS - 1) ? pfo : (3 * NPTS - 1);
                const float* pfp = p + pfo;
                asm volatile("global_prefetch_b8 %0, off" ::"v"(pfp));
            }
            int bB = tbase + 48 + moff;                    // tile pt+1
            b0 = p[bB + 0]; b1 = p[bB + 1]; b2 = p[bB + 2];
            process(pt, a0, a1, a2);                       // consume A, B in flight
            int bA = tbase + 96 + moff;                    // tile pt+2
            a0 = p[bA + 0]; a1 = p[bA + 1]; a2 = p[bA + 2];
            process(pt + 1, b0, b1, b2);                   // consume B, A in flight
            tbase += 96;
        }
        // epilogue: pt == NT-2
        int bB = tbase + 48 + moff;
        b0 = p[bB + 0]; b1 = p[bB + 1]; b2 = p[bB + 2];
        process(NT - 2, a0, a1, a2);
        process(NT - 1, b0, b1, b2);
    }

    // ---- num_neighbors = min(count, K); all 32 lanes write one query ------
    unsigned nv = 0u;
#pragma unroll
    for (int j = 0; j < 16; ++j) {
        unsigned csel = hi ? cnt1[j] : cnt0[j];
        nv = (m == j) ? csel : nv;
    }
    unsigned cc = nv < (unsigned)KNN ? nv : (unsigned)KNN;
    nnOut[(qt0 + hi) * 16 + m] = (int)cc;
}

// ---------------------------------------------------------------------------
// d_out layout (flat, reference return order):
//   mapping        int32  [65536*32]        @ 0
//   num_neighbors  int32  [65536]           @ 2097152
//   outputs        float  [65536*32*3]      @ 2162688
// ---------------------------------------------------------------------------
extern "C" void kernel_launch(void* const* d_in, const int* in_sizes, int n_in,
                              void* d_out, int out_size, void* d_ws, size_t ws_size,
                              hipStream_t stream) {
    const float* q = (const float*)d_in[0];  // points1 [1,65536,3]
    const float* p = (const float*)d_in[1];  // points2 [1,65536,3]

    float* out    = (float*)d_out;
    int*   mapOut = (int*)out;
    int*   nnOut  = mapOut + NPTS * KNN;
    float* outOut = out + NPTS * KNN + NPTS;

    int n4 = out_size / 4;  // out_size = 8454144, divisible by 4
    bq_zero_kernel<<<2048, 256, 0, stream>>>(out, n4);
    // 256 blocks x 8 waves x 32 queries = 65536 queries
    bq_main_kernel<<<NPTS / 256, 256, 0, stream>>>(q, p, mapOut, nnOut, outOut);
}